// Custom_CrossEntropy_PSKD_version9_89601607729502
// MI455X (gfx1250) — compile-verified
//
#include <hip/hip_runtime.h>

#define NCLS   100
#define LPR    25            // active lanes per row: 25 * 4 = 100 elements
#define WPB    8             // waves per block
#define BLOCK  256
#define GRID   2048
#define NWIN   19
#define WINLEN 10
#define ALPHA  0.5f

typedef __attribute__((ext_vector_type(2))) float v2f;
typedef __attribute__((ext_vector_type(8))) float v8f;

__device__ __forceinline__ float wave_sum32(float v) {
#pragma unroll
  for (int d = 16; d >= 1; d >>= 1) v += __shfl_xor(v, d, 32);
  return v;
}
__device__ __forceinline__ float wave_max32(float v) {
#pragma unroll
  for (int d = 16; d >= 1; d >>= 1) v = fmaxf(v, __shfl_xor(v, d, 32));
  return v;
}

__global__ __launch_bounds__(BLOCK) void pskd_main_kernel(
    const float* __restrict__ out, const float* __restrict__ tgt,
    float* __restrict__ partial, int rows) {
  __shared__ __align__(16) float sT[WPB][104];   // raw targets, original order
  __shared__ __align__(16) float sO[WPB][104];   // raw outputs, original order
  __shared__ unsigned sIdx[WPB][32];             // rank -> orig index, 4 bytes/lane
  __shared__ float blockAcc[WPB];

  const int lane   = threadIdx.x & 31;
  const int w      = threadIdx.x >> 5;
  const int waveId = blockIdx.x * WPB + w;
  const int nWaves = GRID * WPB;
  const bool act   = (lane < LPR);
  const float NEG_INF = -__builtin_inff();

  float acc = 0.0f;

  for (int row = waveId; row < rows; row += nWaves) {
    const float* orow = out + (size_t)row * NCLS;
    const float* trow = tgt + (size_t)row * NCLS;

    // ---- coalesced b128 load: element e = 4*lane + s ----
    float4 o4 = make_float4(0.f, 0.f, 0.f, 0.f);
    float4 t4 = make_float4(0.f, 0.f, 0.f, 0.f);
    if (act) {
      o4 = *(reinterpret_cast<const float4*>(orow) + lane);
      t4 = *(reinterpret_cast<const float4*>(trow) + lane);
      // prefetch next row (global_prefetch_b8)
      const int nrow = row + nWaves;
      if (nrow < rows) {
        __builtin_prefetch(out + (size_t)nrow * NCLS + 4 * lane, 0, 1);
        __builtin_prefetch(tgt + (size_t)nrow * NCLS + 4 * lane, 0, 1);
      }
      // stage raw row to LDS in original order (exact values for windows)
      *(reinterpret_cast<float4*>(&sT[w][0]) + lane) = t4;
      *(reinterpret_cast<float4*>(&sO[w][0]) + lane) = o4;
    }

    // ---- main CE: sum(-t * logsoftmax(o)) = st*logZ - dot ----
    float mo = NEG_INF;
    if (act) mo = fmaxf(fmaxf(o4.x, o4.y), fmaxf(o4.z, o4.w));
    mo = wave_max32(mo);
    float se = 0.f, st = 0.f, dot = 0.f;
    if (act) {
      se  = __expf(o4.x - mo) + __expf(o4.y - mo) +
            __expf(o4.z - mo) + __expf(o4.w - mo);
      st  = t4.x + t4.y + t4.z + t4.w;
      dot = t4.x * o4.x + t4.y * o4.y + t4.z * o4.z + t4.w * o4.w;
    }
    se  = wave_sum32(se);
    st  = wave_sum32(st);
    dot = wave_sum32(dot);
    const float rowLoss = st * (mo + __logf(se)) - dot;

    // ---- index-packed keys: bits(t) | (127 - e) in low 7 bits ----
    // targets are positive floats -> uint order == float order; packing gives
    // stable (argsort-matching) tie-break; pads (e >= 100) sink to the end.
    unsigned pk[4];
    {
      const unsigned e0 = (unsigned)lane << 2;
      if (act) {
        pk[0] = (__float_as_uint(t4.x) & 0xFFFFFF80u) | (127u - (e0 + 0u));
        pk[1] = (__float_as_uint(t4.y) & 0xFFFFFF80u) | (127u - (e0 + 1u));
        pk[2] = (__float_as_uint(t4.z) & 0xFFFFFF80u) | (127u - (e0 + 2u));
        pk[3] = (__float_as_uint(t4.w) & 0xFFFFFF80u) | (127u - (e0 + 3u));
      } else {
#pragma unroll
        for (int s = 0; s < 4; ++s) pk[s] = 127u - (e0 + (unsigned)s);
      }
    }

    // ---- wave-parallel bitonic sort of 128 u32 keys, descending ----
#pragma unroll
    for (int kk = 1; kk <= 7; ++kk) {
      const unsigned ksz = 1u << kk;
#pragma unroll
      for (int jj = kk - 1; jj >= 0; --jj) {
        if (jj >= 2) {
          // cross-lane stage: lane xor = 2^(jj-2) in {1,2,4,8,16}
          const int lx = 1 << (jj - 2);
          const bool isLow = ((lane & lx) == 0);
          const bool desc  = ((((unsigned)lane << 2) & ksz) == 0u);
          const bool keepMax = (isLow == desc);
#pragma unroll
          for (int s = 0; s < 4; ++s) {
            const unsigned p = (unsigned)__shfl_xor((int)pk[s], lx, 32);
            const bool take = keepMax ? (p > pk[s]) : (p < pk[s]);
            pk[s] = take ? p : pk[s];
          }
        } else {
          // in-register stage: j=1 -> (0,1),(2,3); j=2 -> (0,2),(1,3)
          const int j = 1 << jj;
#pragma unroll
          for (int p2 = 0; p2 < 2; ++p2) {
            const int a = (jj == 1) ? p2 : 2 * p2;
            const int b = a + j;
            const unsigned i = ((unsigned)lane << 2) + (unsigned)a;
            const bool desc = ((i & ksz) == 0u);
            const unsigned hi = pk[a] > pk[b] ? pk[a] : pk[b];
            const unsigned lo = pk[a] > pk[b] ? pk[b] : pk[a];
            pk[a] = desc ? hi : lo;
            pk[b] = desc ? lo : hi;
          }
        }
      }
    }

    // ---- store rank->orig permutation (4 ranks packed per lane) ----
    {
      const unsigned g0 = 127u - (pk[0] & 127u);
      const unsigned g1 = 127u - (pk[1] & 127u);
      const unsigned g2 = 127u - (pk[2] & 127u);
      const unsigned g3 = 127u - (pk[3] & 127u);
      sIdx[w][lane] = g0 | (g1 << 8) | (g2 << 16) | (g3 << 24);
    }
    __threadfence_block();  // wave-internal RAW across lanes

    // ---- 19 rank windows, one per lane; indirect LDS gather ----
    float sub = 0.0f;
    if (lane < NWIN) {
      const int s0 = 5 * lane;
      float tb[WINLEN], ob[WINLEN];
      float mt = NEG_INF, mo2 = NEG_INF;
#pragma unroll
      for (int i = 0; i < WINLEN; ++i) {
        const int r = s0 + i;                       // rank < 100
        const unsigned pw = sIdx[w][r >> 2];
        const unsigned og = (pw >> ((r & 3) * 8)) & 127u;
        tb[i] = sT[w][og];
        ob[i] = sO[w][og];
        mt  = fmaxf(mt,  tb[i]);
        mo2 = fmaxf(mo2, ob[i]);
      }
      float set = 0.f, seo = 0.f, num = 0.f;
#pragma unroll
      for (int i = 0; i < WINLEN; ++i) {
        const float et = __expf(tb[i] - mt);
        set += et;
        num += et * ob[i];
        seo += __expf(ob[i] - mo2);
      }
      // sum_i softmax(t)_i * (logZ_o - o_i) = logZ_o - num/set
      sub = (mo2 + __logf(seo)) - num / set;
    }
    sub = wave_sum32(sub);

    acc += rowLoss + ALPHA * sub;
  }

  if (lane == 0) blockAcc[w] = acc;
  __syncthreads();
  if (threadIdx.x == 0) {
    float s = 0.f;
#pragma unroll
    for (int i = 0; i < WPB; ++i) s += blockAcc[i];
    partial[blockIdx.x] = s;
  }
}

// Deterministic final reduce: 1 wave; 32 lane-sums collapsed with
// v_wmma_f32_16x16x4_f32 (B = ones => D rows are A row-sums).
__global__ __launch_bounds__(32) void pskd_reduce_kernel(
    const float* __restrict__ partial, float* __restrict__ result,
    int n, float invBatch) {
  const int lane = threadIdx.x;
  float s = 0.f;
  for (int i = lane; i < n; i += 32) s += partial[i];

#if __has_builtin(__builtin_amdgcn_wmma_f32_16x16x4_f32)
  // A (16x4 f32, 2 VGPRs): lane m -> A[m][0], lane m+16 -> A[m][2]; rest 0.
  v2f a; a.x = s;   a.y = 0.f;
  v2f b; b.x = 1.f; b.y = 1.f;     // B (4x16) = all ones
  v8f c = {};
  v8f d = __builtin_amdgcn_wmma_f32_16x16x4_f32(
      /*neg_a=*/false, a, /*neg_b=*/false, b,
      /*c_mod=*/(short)0, c, /*reuse_a=*/false, /*reuse_b=*/false);
  // lanes 0-15: d[0..7] = rowsums M=0..7; lanes 16-31: M=8..15.
  float t = 0.f;
#pragma unroll
  for (int r = 0; r < 8; ++r) t += d[r];
  t += __shfl_xor(t, 16, 32);
#else
  float t = s;
#pragma unroll
  for (int dd = 16; dd >= 1; dd >>= 1) t += __shfl_xor(t, dd, 32);
#endif

  if (lane == 0) result[0] = t * invBatch;
}

extern "C" void kernel_launch(void* const* d_in, const int* in_sizes, int n_in,
                              void* d_out, int out_size, void* d_ws, size_t ws_size,
                              hipStream_t stream) {
  const float* output  = (const float*)d_in[0];
  const float* targets = (const float*)d_in[1];
  const int rows = in_sizes[0] / NCLS;     // BATCH
  float* partial = (float*)d_ws;           // GRID floats of scratch

  pskd_main_kernel<<<GRID, BLOCK, 0, stream>>>(output, targets, partial, rows);
  pskd_reduce_kernel<<<1, 32, 0, stream>>>(partial, (float*)d_out, GRID,
                                           1.0f / (float)rows);
}